// PointSectorScatter_40742059770602
// MI455X (gfx1250) — compile-verified
//
#include <hip/hip_runtime.h>
#include <stdint.h>

// Problem constants (from the reference): C=32 channels, 64x256 canvas, B=64.
#define C_CH   32
#define NH_    64
#define NW_    256
#define HW_    (NH_ * NW_)        // 16384 = 2^14
#define CHW_   (C_CH * HW_)       // 524288 = 2^19

// Clang vector type: usable with __builtin_nontemporal_load (HIP float4 is not).
typedef float f32x4 __attribute__((ext_vector_type(4)));

// ---------------------------------------------------------------------------
// Kernel 1: clear the occupancy mask (1 byte per (b,h,w) cell), 16B stores.
// ---------------------------------------------------------------------------
__global__ __launch_bounds__(256)
void pss_clear_mask(uint4* __restrict__ mask16, int n16) {
    int i = blockIdx.x * blockDim.x + threadIdx.x;
    if (i < n16) mask16[i] = make_uint4(0u, 0u, 0u, 0u);
}

// ---------------------------------------------------------------------------
// Kernel 2: mark occupied cells. Cells are unique (reference guarantees no
// duplicate scatter indices), so plain byte stores — no atomics needed.
// Coords are streamed once -> non-temporal loads (keep L2 for the canvas).
// ---------------------------------------------------------------------------
__global__ __launch_bounds__(256)
void pss_build_mask(const int* __restrict__ coords, uint8_t* __restrict__ mask, int n) {
    int i = blockIdx.x * blockDim.x + threadIdx.x;
    if (i >= n) return;
    const int* cc = coords + 3 * i;
    int b = __builtin_nontemporal_load(cc + 0);
    int h = __builtin_nontemporal_load(cc + 1);
    int w = __builtin_nontemporal_load(cc + 2);
    unsigned cell = ((unsigned)b << 14) + ((unsigned)h << 8) + (unsigned)w;
    mask[cell] = (uint8_t)1;
}

// ---------------------------------------------------------------------------
// Kernel 3: zero-fill canvas with float4 stores, skipping quads whose 4 cells
// are all occupied (they will be fully overwritten by the scatter). Regular-
// temporal stores on purpose: keep the 128 MiB canvas resident in the 192 MB
// L2 so the scatter's 4B writes hit L2 lines instead of RMW-ing HBM lines.
// dword index V -> b = V>>19, pos-in-image = V & 16383 (independent of c).
// ---------------------------------------------------------------------------
__global__ __launch_bounds__(256)
void pss_zero_masked(float4* __restrict__ out, const uint8_t* __restrict__ mask, int nvec) {
    int i = blockIdx.x * blockDim.x + threadIdx.x;
    if (i >= nvec) return;
    unsigned V    = ((unsigned)i) << 2;                       // dword index, w%4==0
    unsigned cell = ((V >> 19) << 14) | (V & 16383u);         // b*HW + h*NW + w
    unsigned m = *reinterpret_cast<const unsigned*>(mask + cell);  // 4-byte aligned
    if (m == 0x01010101u) return;                             // all 4 cells occupied
    out[i] = make_float4(0.f, 0.f, 0.f, 0.f);
}

// Fallback: unconditional zero-fill (used if d_ws is too small for the mask).
__global__ __launch_bounds__(256)
void pss_zero_plain(float4* __restrict__ out, int nvec) {
    int i = blockIdx.x * blockDim.x + threadIdx.x;
    if (i < nvec) out[i] = make_float4(0.f, 0.f, 0.f, 0.f);
}

// ---------------------------------------------------------------------------
// Kernel 4a: vectorized scatter — 4 points per lane (requires n % 4 == 0).
// Feature loads become global_load_b128 th:NT (coalesced across the wave32,
// 16B-aligned since n % 4 == 0 and lane handles points 4q..4q+3). The 32x4
// scattered 4B channel stores hit the L2-resident canvas lines written by the
// zero-fill. This cuts VMEM instruction issue in the hot kernel by ~36%.
// ---------------------------------------------------------------------------
__global__ __launch_bounds__(256)
void pss_scatter_x4(const float* __restrict__ features,
                    const int*   __restrict__ coords,
                    float*       __restrict__ out, int n) {
    int q = blockIdx.x * blockDim.x + threadIdx.x;   // quad-of-points index
    int i0 = q * 4;
    if (i0 >= n) return;

    unsigned base[4];
#pragma unroll
    for (int k = 0; k < 4; ++k) {
        const int* cc = coords + 3 * (i0 + k);
        int b = __builtin_nontemporal_load(cc + 0);
        int h = __builtin_nontemporal_load(cc + 1);
        int w = __builtin_nontemporal_load(cc + 2);
        base[k] = ((unsigned)b * (unsigned)CHW_) + ((unsigned)h << 8) + (unsigned)w;
    }

#pragma unroll
    for (int c = 0; c < C_CH; ++c) {
        const f32x4* src =
            reinterpret_cast<const f32x4*>(features + (size_t)c * (size_t)n + (size_t)i0);
        f32x4 v = __builtin_nontemporal_load(src);
        unsigned coff = (unsigned)c * (unsigned)HW_;
        out[base[0] + coff] = v.x;
        out[base[1] + coff] = v.y;
        out[base[2] + coff] = v.z;
        out[base[3] + coff] = v.w;
    }
}

// ---------------------------------------------------------------------------
// Kernel 4b: scalar scatter fallback (any n). One point per lane.
// ---------------------------------------------------------------------------
__global__ __launch_bounds__(256)
void pss_scatter(const float* __restrict__ features,
                 const int*   __restrict__ coords,
                 float*       __restrict__ out, int n) {
    int i = blockIdx.x * blockDim.x + threadIdx.x;
    if (i >= n) return;
    const int* cc = coords + 3 * i;
    int b = __builtin_nontemporal_load(cc + 0);
    int h = __builtin_nontemporal_load(cc + 1);
    int w = __builtin_nontemporal_load(cc + 2);
    unsigned base = ((unsigned)b * (unsigned)CHW_) + ((unsigned)h << 8) + (unsigned)w;
#pragma unroll
    for (int c = 0; c < C_CH; ++c) {
        float v = __builtin_nontemporal_load(features + (size_t)c * (size_t)n + (size_t)i);
        out[base + (unsigned)c * (unsigned)HW_] = v;
    }
}

// ---------------------------------------------------------------------------
// Host launcher. Inputs: d_in[0]=features (f32, C*N), d_in[1]=coords (i32, N*3),
// d_in[2]=batch_size (i32 scalar, unused: encoded in out_size). Output: f32.
// ---------------------------------------------------------------------------
extern "C" void kernel_launch(void* const* d_in, const int* in_sizes, int n_in,
                              void* d_out, int out_size, void* d_ws, size_t ws_size,
                              hipStream_t stream) {
    const float* features = (const float*)d_in[0];
    const int*   coords   = (const int*)d_in[1];
    float*       out      = (float*)d_out;

    const int n    = in_sizes[1] / 3;     // number of points (900,000)
    const int nvec = out_size / 4;        // float4 elements in canvas (2^23)

    // mask bytes = one per (b,h,w) cell = out_size / C  (B*HW)
    const size_t mask_bytes = (size_t)(out_size / C_CH);

    const int TPB = 256;

    if (ws_size >= mask_bytes && (mask_bytes & 15u) == 0) {
        uint8_t* mask = (uint8_t*)d_ws;
        const int n16 = (int)(mask_bytes / 16);

        pss_clear_mask <<<(n16  + TPB - 1) / TPB, TPB, 0, stream>>>((uint4*)mask, n16);
        pss_build_mask <<<(n    + TPB - 1) / TPB, TPB, 0, stream>>>(coords, mask, n);
        pss_zero_masked<<<(nvec + TPB - 1) / TPB, TPB, 0, stream>>>((float4*)out, mask, nvec);
    } else {
        pss_zero_plain <<<(nvec + TPB - 1) / TPB, TPB, 0, stream>>>((float4*)out, nvec);
    }

    if ((n & 3) == 0) {
        const int nq = n / 4;
        pss_scatter_x4<<<(nq + TPB - 1) / TPB, TPB, 0, stream>>>(features, coords, out, n);
    } else {
        pss_scatter   <<<(n  + TPB - 1) / TPB, TPB, 0, stream>>>(features, coords, out, n);
    }
}